// ClusterisedLinearNetwork_47107201303341
// MI455X (gfx1250) — compile-verified
//
#include <hip/hip_runtime.h>
#include <hip/hip_bf16.h>

// ---------------------------------------------------------------------------
// ClusterisedLinearNetwork for MI455X (gfx1250, wave32).
//
// Roofline: per token we need enc(120) . 9 gathered W rows = ~1080 MACs +
// 120 sin/cos; total ~0.57 GFLOP, ~15MB traffic -> compute-bound on gathered
// dots. We bin (token,k) pairs by cluster id so each bin shares 3 W rows,
// turning the gather into a real 16x120 @ 120x3 GEMM per tile that runs on
// V_WMMA_F32_16X16X4_F32. B fragments (shared W rows) are preloaded from a
// zero-padded LDS image once per chunk and held in VGPRs, so the tile loop is
// pure trans + WMMA issue.
// ---------------------------------------------------------------------------

typedef __attribute__((ext_vector_type(2))) float v2f;
typedef __attribute__((ext_vector_type(8))) float v8f;

#define NCLUST 256
#define DFEAT  120      // 6 coords * 2 (sin,cos) * 10 freqs
#define NSTEP  30       // K=120 in steps of 4
#define BINCAP 4096     // mean bin size 3072, sigma ~55 -> 4096 is >16 sigma
#define NCHUNK 6

__device__ __forceinline__ v8f wmma_f32_16x16x4(v2f a, v2f b, v8f c) {
#if defined(__has_builtin) && __has_builtin(__builtin_amdgcn_wmma_f32_16x16x4_f32)
  // 8 args per probe: (neg_a, A, neg_b, B, c_mod, C, reuse_a, reuse_b)
  return __builtin_amdgcn_wmma_f32_16x16x4_f32(false, a, false, b, (short)0, c,
                                               false, false);
#else
  asm volatile("v_wmma_f32_16x16x4_f32 %0, %1, %2, %0"
               : "+v"(c) : "v"(a), "v"(b));
  return c;
#endif
}

// ---------------------------------------------------------------------------
// K1: bin (n,k) pairs by cluster id with LDS-aggregated counting scatter.
// cursors[c] ends up holding the bin count. Entry = (n<<2)|k.
// ---------------------------------------------------------------------------
__global__ void __launch_bounds__(256)
bin_scatter_kernel(const int* __restrict__ ids, unsigned* __restrict__ cursors,
                   unsigned* __restrict__ entries, int Ntok) {
  __shared__ unsigned hist[NCLUST];
  __shared__ unsigned basev[NCLUST];
  __shared__ unsigned h2[NCLUST];
  const int tid = threadIdx.x;
  hist[tid] = 0u;
  __syncthreads();

  const int n0 = blockIdx.x * 512 + tid;
#pragma unroll
  for (int r = 0; r < 2; ++r) {
    const int n = n0 + r * 256;
    if (n < Ntok) {
#pragma unroll
      for (int k = 0; k < 3; ++k) {
        const int cid = ids[n * 3 + k] & (NCLUST - 1);
        atomicAdd(&hist[cid], 1u);
      }
    }
  }
  __syncthreads();
  basev[tid] = atomicAdd(&cursors[tid], hist[tid]);  // reserve block in bin
  h2[tid] = 0u;
  __syncthreads();
#pragma unroll
  for (int r = 0; r < 2; ++r) {
    const int n = n0 + r * 256;
    if (n < Ntok) {
#pragma unroll
      for (int k = 0; k < 3; ++k) {
        const int cid = ids[n * 3 + k] & (NCLUST - 1);
        const unsigned slot = atomicAdd(&h2[cid], 1u);
        const unsigned pos = basev[cid] + slot;
        if (pos < BINCAP)
          entries[(size_t)cid * BINCAP + pos] = ((unsigned)n << 2) | (unsigned)k;
      }
    }
  }
}

// ---------------------------------------------------------------------------
// K2: per-bin WMMA GEMM. One tile = 16 entries (same cluster). A-fragment is
// the positional encoding computed on the fly (sin/cos co-execute with the
// matrix pipe); B-fragments are preloaded into VGPRs from a zero-padded LDS
// image of the 3 shared W rows; accumulate over K=120 in 30 WMMA steps.
// dots[n][k][c] = weights[k][n] * (enc . Wrow).
// ---------------------------------------------------------------------------
__global__ void __launch_bounds__(128)
bin_wmma_kernel(const float* __restrict__ X, const float* __restrict__ W,
                const float* __restrict__ weights,
                const unsigned* __restrict__ cursors,
                const unsigned* __restrict__ entries,
                float* __restrict__ dots, int Ntok) {
  // 16 rows so every B lane (N=0..15) has a valid source; rows 3..15 are zero.
  __shared__ float ldsW[16 * DFEAT];
  const int bin = blockIdx.x;
  const int chunk = blockIdx.y;
  const int tid = threadIdx.x;

  for (int i = tid; i < 16 * DFEAT; i += 128)
    ldsW[i] = (i < 3 * DFEAT) ? W[(size_t)bin * (3 * DFEAT) + i] : 0.f;
  __syncthreads();

  int count = (int)cursors[bin];
  if (count > BINCAP) count = BINCAP;
  const int per = (count + NCHUNK - 1) / NCHUNK;
  const int start = chunk * per;
  const int end = min(start + per, count);
  const int cc = end - start;
  if (cc <= 0) return;

  const int wave = tid >> 5;
  const int lane = tid & 31;
  const int mlo = lane & 15;     // token slot within tile (both half-waves)
  const bool hi = lane >= 16;    // half-wave selects K phase {2,3} vs {0,1}
  const int boff = hi ? 2 : 0;

  // Hoist all B fragments: B VGPR0 = K{4s+0 | 4s+2}, VGPR1 = K{4s+1 | 4s+3}.
  // Index is even -> single aligned ds_load_b64 per step, no predication.
  v2f bfrag[NSTEP];
  {
    const float* wrow = &ldsW[mlo * DFEAT + boff];
#pragma unroll
    for (int s = 0; s < NSTEP; ++s)
      bfrag[s] = *(const v2f*)(wrow + 4 * s);
  }

  const int ntiles = (cc + 15) >> 4;
  for (int t = wave; t < ntiles; t += 4) {
    const int tstart = start + t * 16;
    const int valid = min(16, end - tstart);
    const size_t ebase = (size_t)bin * BINCAP + tstart;

    const unsigned e = entries[ebase + (mlo < valid ? mlo : 0)];
    const int n = (int)(e >> 2);
    const float* xp = X + (size_t)n * 6;
    const float x0 = xp[0], x1 = xp[1], x2 = xp[2];
    const float x3 = xp[3], x4 = xp[4], x5 = xp[5];

    v8f acc = {0.f, 0.f, 0.f, 0.f, 0.f, 0.f, 0.f, 0.f};

#pragma unroll
    for (int f = 0; f < 10; ++f) {
      const float fr = (float)(1 << f);
      // enc index = f*12 + j : j<6 -> sin(x_j*2^f), j>=6 -> cos(x_{j-6}*2^f)
      const float s0 = __sinf(x0 * fr), s1 = __sinf(x1 * fr);
      const float s2 = __sinf(x2 * fr), s3 = __sinf(x3 * fr);
      const float s4 = __sinf(x4 * fr), s5 = __sinf(x5 * fr);
      const float c0 = __cosf(x0 * fr), c1 = __cosf(x1 * fr);
      const float c2 = __cosf(x2 * fr), c3 = __cosf(x3 * fr);
      const float c4 = __cosf(x4 * fr), c5 = __cosf(x5 * fr);
      v2f a;
      // step 0: K = 12f + {0,1} (lanes 0-15) / {2,3} (lanes 16-31)
      a.x = hi ? s2 : s0; a.y = hi ? s3 : s1;
      acc = wmma_f32_16x16x4(a, bfrag[f * 3 + 0], acc);
      // step 1: K = 12f+4 + phase
      a.x = hi ? c0 : s4; a.y = hi ? c1 : s5;
      acc = wmma_f32_16x16x4(a, bfrag[f * 3 + 1], acc);
      // step 2: K = 12f+8 + phase
      a.x = hi ? c4 : c2; a.y = hi ? c5 : c3;
      acc = wmma_f32_16x16x4(a, bfrag[f * 3 + 2], acc);
    }

    // D layout: VGPR r holds row M=r (lanes 0-15, N=lane) and M=r+8
    // (lanes 16-31, N=lane-16). Live channels are N=0..2.
    int ch = -1;
    if (lane < 3) ch = lane;
    else if (lane >= 16 && lane < 19) ch = lane - 16;
    if (ch >= 0) {
      const int mofs = hi ? 8 : 0;
#pragma unroll
      for (int m = 0; m < 8; ++m) {
        const int mg = m + mofs;
        if (mg < valid) {
          const unsigned ee = entries[ebase + mg];
          const int nn = (int)(ee >> 2);
          const int kk = (int)(ee & 3u);
          const float wv = weights[(size_t)kk * Ntok + nn];
          dots[((size_t)nn * 3 + kk) * 3 + ch] = wv * acc[m];
        }
      }
    }
  }
}

// ---------------------------------------------------------------------------
// K3: combine the 3 weighted cluster dots per token, apply the (-1,-1,-1) mask.
// ---------------------------------------------------------------------------
__global__ void __launch_bounds__(256)
combine_kernel(const float* __restrict__ X, const float* __restrict__ dots,
               float* __restrict__ out, int Ntok) {
  const int n = blockIdx.x * 256 + threadIdx.x;
  if (n >= Ntok) return;
  const float* xp = X + (size_t)n * 6;
  const bool mask = (xp[0] == -1.0f) && (xp[1] == -1.0f) && (xp[2] == -1.0f);
  const float* dp = dots + (size_t)n * 9;
  const float r0 = dp[0] + dp[3] + dp[6];
  const float r1 = dp[1] + dp[4] + dp[7];
  const float r2 = dp[2] + dp[5] + dp[8];
  out[(size_t)n * 3 + 0] = mask ? 0.f : r0;
  out[(size_t)n * 3 + 1] = mask ? 0.f : r1;
  out[(size_t)n * 3 + 2] = mask ? 0.f : r2;
}

// ---------------------------------------------------------------------------
// Fallback (only if ws too small): straight per-token gathered dots on VALU.
// ---------------------------------------------------------------------------
__global__ void __launch_bounds__(256)
naive_kernel(const float* __restrict__ X, const float* __restrict__ W,
             const float* __restrict__ weights, const int* __restrict__ ids,
             float* __restrict__ out, int Ntok) {
  const int n = blockIdx.x * 256 + threadIdx.x;
  if (n >= Ntok) return;
  const float* xp = X + (size_t)n * 6;
  float xv[6];
#pragma unroll
  for (int j = 0; j < 6; ++j) xv[j] = xp[j];
  const bool mask = (xv[0] == -1.0f) && (xv[1] == -1.0f) && (xv[2] == -1.0f);

  float enc[DFEAT];
#pragma unroll
  for (int f = 0; f < 10; ++f) {
    const float fr = (float)(1 << f);
#pragma unroll
    for (int j = 0; j < 6; ++j) {
      enc[f * 12 + j] = __sinf(xv[j] * fr);
      enc[f * 12 + 6 + j] = __cosf(xv[j] * fr);
    }
  }
  float r[3] = {0.f, 0.f, 0.f};
#pragma unroll
  for (int k = 0; k < 3; ++k) {
    const int cid = ids[n * 3 + k] & (NCLUST - 1);
    const float wv = weights[(size_t)k * Ntok + n];
#pragma unroll
    for (int c = 0; c < 3; ++c) {
      const float* row = W + ((size_t)cid * 3 + c) * DFEAT;
      float acc = 0.f;
#pragma unroll
      for (int j = 0; j < DFEAT; ++j) acc = fmaf(enc[j], row[j], acc);
      r[c] = fmaf(wv, acc, r[c]);
    }
  }
#pragma unroll
  for (int c = 0; c < 3; ++c) out[(size_t)n * 3 + c] = mask ? 0.f : r[c];
}

extern "C" void kernel_launch(void* const* d_in, const int* in_sizes, int n_in,
                              void* d_out, int out_size, void* d_ws,
                              size_t ws_size, hipStream_t stream) {
  const float* X = (const float*)d_in[0];
  const float* W = (const float*)d_in[1];
  const float* weights = (const float*)d_in[2];
  const int* ids = (const int*)d_in[3];
  float* out = (float*)d_out;
  const int Ntok = in_sizes[0] / 6;

  const size_t off_entries = 1024;
  const size_t off_dots = off_entries + (size_t)NCLUST * BINCAP * sizeof(unsigned);
  const size_t need = off_dots + (size_t)Ntok * 9 * sizeof(float);

  if (ws_size >= need) {
    unsigned char* ws = (unsigned char*)d_ws;
    unsigned* cursors = (unsigned*)ws;
    unsigned* entries = (unsigned*)(ws + off_entries);
    float* dots = (float*)(ws + off_dots);

    hipMemsetAsync(cursors, 0, NCLUST * sizeof(unsigned), stream);
    hipMemsetAsync(dots, 0, (size_t)Ntok * 9 * sizeof(float), stream);

    const int nwg = (Ntok + 511) / 512;
    bin_scatter_kernel<<<nwg, 256, 0, stream>>>(ids, cursors, entries, Ntok);
    bin_wmma_kernel<<<dim3(NCLUST, NCHUNK), 128, 0, stream>>>(
        X, W, weights, cursors, entries, dots, Ntok);
    combine_kernel<<<(Ntok + 255) / 256, 256, 0, stream>>>(X, dots, out, Ntok);
  } else {
    naive_kernel<<<(Ntok + 255) / 256, 256, 0, stream>>>(X, W, weights, ids,
                                                         out, Ntok);
  }
}